// GCNFirstLayer_10660108829138
// MI455X (gfx1250) — compile-verified
//
#include <hip/hip_runtime.h>

// Problem constants (match reference)
#define N_NODES 10000
#define N_EDGES 320000
#define F_IN    256
#define F_OUT   256
#define K_DIM   512      // 2*F_IN
#define EPB     16       // edges per block in scatter kernel
#define LDSTR   516      // padded LDS row stride (floats) -> conflict-free

typedef __attribute__((ext_vector_type(2))) float v2f;
typedef __attribute__((ext_vector_type(8))) float v8f;

// ---------------------------------------------------------------------------
// Zero the workspace accumulators (sums [N*F] + cnt [N]), float4 stores.
// ---------------------------------------------------------------------------
__global__ void zero_ws_kernel(float4* __restrict__ p, int n4) {
  int i = blockIdx.x * blockDim.x + threadIdx.x;
  if (i < n4) p[i] = make_float4(0.f, 0.f, 0.f, 0.f);
}

// ---------------------------------------------------------------------------
// Edge message + scatter-add.
// 256 threads/block, 64 threads per edge (float4 each = 256 features),
// 4 edges in flight, EPB edges per block. feature & sums are L2-resident,
// so the random gathers and f32 atomics never touch HBM.
// ---------------------------------------------------------------------------
__global__ __launch_bounds__(256) void edge_scatter_kernel(
    const float* __restrict__ feat, const float* __restrict__ ew,
    const int* __restrict__ src, const int* __restrict__ dst,
    float* __restrict__ sums, float* __restrict__ cnt)
{
  const int tid = threadIdx.x;
  const int li  = tid & 63;   // chunk index within an edge row (64 x float4)
  const int eg  = tid >> 6;   // 0..3: which of 4 concurrent edges

  // Broadcast row feature[src[0]] -- same chunk for every edge this thread sees
  const int s0 = src[0];
  const float4 h0 = ((const float4*)(feat + (size_t)s0 * F_IN))[li];

  const int ebase = blockIdx.x * EPB;
#pragma unroll
  for (int it = 0; it < EPB / 4; ++it) {
    const int e = ebase + it * 4 + eg;          // grid sized so e < N_EDGES
    const int s = src[e];
    const int d = dst[e];
    const float w = ew[e];

    const float4 hs = ((const float4*)(feat + (size_t)s * F_IN))[li];
    float4 m;
    m.x = h0.x * hs.x * w;
    m.y = h0.y * hs.y * w;
    m.z = h0.z * hs.z * w;
    m.w = h0.w * hs.w * w;

    float* o = sums + (size_t)d * F_IN + li * 4;
    __hip_atomic_fetch_add(o + 0, m.x, __ATOMIC_RELAXED, __HIP_MEMORY_SCOPE_AGENT);
    __hip_atomic_fetch_add(o + 1, m.y, __ATOMIC_RELAXED, __HIP_MEMORY_SCOPE_AGENT);
    __hip_atomic_fetch_add(o + 2, m.z, __ATOMIC_RELAXED, __HIP_MEMORY_SCOPE_AGENT);
    __hip_atomic_fetch_add(o + 3, m.w, __ATOMIC_RELAXED, __HIP_MEMORY_SCOPE_AGENT);

    if (li == 0)
      __hip_atomic_fetch_add(cnt + d, 1.0f, __ATOMIC_RELAXED, __HIP_MEMORY_SCOPE_AGENT);
  }
}

// ---------------------------------------------------------------------------
// In-place mean: sums[n][*] /= max(cnt[n],1)
// ---------------------------------------------------------------------------
__global__ void mean_kernel(float4* __restrict__ sums4, const float* __restrict__ cnt) {
  int i = blockIdx.x * blockDim.x + threadIdx.x;  // over N_NODES * (F_IN/4)
  if (i >= N_NODES * (F_IN / 4)) return;
  const int n = i >> 6;                            // /(F_IN/4)
  const float inv = 1.0f / fmaxf(cnt[n], 1.0f);
  float4 v = sums4[i];
  v.x *= inv; v.y *= inv; v.z *= inv; v.w *= inv;
  sums4[i] = v;
}

// ---------------------------------------------------------------------------
// out = relu([feature | mean] @ W^T + b) via V_WMMA_F32_16X16X4_F32.
// Block = 512 threads = 16 waves. Block owns one 16-row M tile; each wave
// owns one 16-col N tile. A tile (16x512 f32) staged into LDS with
// GLOBAL_LOAD_ASYNC_TO_LDS_B128 (ASYNCcnt path, no VGPR staging);
// B read as per-lane float2 from W (L2-resident).
// ---------------------------------------------------------------------------
__global__ __launch_bounds__(512) void gemm_wmma_kernel(
    const float* __restrict__ feat, const float* __restrict__ mean,
    const float* __restrict__ W, const float* __restrict__ bias,
    float* __restrict__ out)
{
  __shared__ float lA[16 * LDSTR];   // sole LDS object -> group-segment offset 0

  const int tid = threadIdx.x;
  const int m0  = blockIdx.x * 16;

  // Async A-tile load: 16 rows x 512 cols (first 256 from feature, last 256
  // from mean). 2048 x b128 transfers, 4 per thread, direct global->LDS.
  // NOTE: the LDS address is derived FROM the lA pointer (ptrtoint) so that
  // lA escapes into the asm; otherwise AA deletes the subsequent ds_loads.
  for (int idx = tid; idx < 16 * (K_DIM / 4); idx += 512) {
    const int row = idx >> 7;         // / (K_DIM/4)
    const int k   = (idx & 127) * 4;
    const float* sp = (k < F_IN)
        ? (feat + (size_t)(m0 + row) * F_IN + k)
        : (mean + (size_t)(m0 + row) * F_IN + (k - F_IN));
    float* lp = &lA[row * LDSTR + k];
    const unsigned lds_b = (unsigned)(uintptr_t)lp;   // low 32 bits = LDS byte addr
    const unsigned long long ga = (unsigned long long)(uintptr_t)sp;
    asm volatile("global_load_async_to_lds_b128 %0, %1, off"
                 :: "v"(lds_b), "v"(ga)
                 : "memory");
  }
  asm volatile("s_wait_asynccnt 0x0" ::: "memory");  // our LDS writes landed
  __syncthreads();                                    // everyone's landed

  const int wave = tid >> 5;          // 0..15 -> N tile
  const int lane = tid & 31;
  const int nl   = lane & 15;         // N (and A-row M) within tile
  const int hi   = lane >> 4;         // 0: K=0,1  1: K=2,3  (per ISA layout)
  const int n0   = wave * 16;

  const float* wrow = W + (size_t)(n0 + nl) * K_DIM;  // W[n][k], k contiguous

  v8f c = {};
  for (int kb = 0; kb < K_DIM / 4; ++kb) {
    const int kk = kb * 4 + hi * 2;
    const v2f a = *(const v2f*)&lA[nl * LDSTR + kk];  // A[M=nl][kk,kk+1]
    const v2f b = *(const v2f*)(wrow + kk);           // B[kk,kk+1][N=n0+nl]
    c = __builtin_amdgcn_wmma_f32_16x16x4_f32(
        /*neg_a=*/false, a, /*neg_b=*/false, b,
        /*c_mod=*/(short)0, c, /*reuse_a=*/false, /*reuse_b=*/false);
  }

  // Epilogue: +bias, relu, store. C layout: VGPR r -> M = r + 8*hi, N = nl.
  const float bv = bias[n0 + nl];
#pragma unroll
  for (int r = 0; r < 8; ++r) {
    const int m = m0 + r + 8 * hi;
    float v = c[r] + bv;
    v = fmaxf(v, 0.0f);
    out[(size_t)m * F_OUT + (n0 + nl)] = v;
  }
}

// ---------------------------------------------------------------------------
extern "C" void kernel_launch(void* const* d_in, const int* in_sizes, int n_in,
                              void* d_out, int out_size, void* d_ws, size_t ws_size,
                              hipStream_t stream) {
  const float* feat = (const float*)d_in[0];
  const float* ew   = (const float*)d_in[1];
  const int*   src  = (const int*)d_in[2];
  const int*   dst  = (const int*)d_in[3];
  const float* W    = (const float*)d_in[4];
  const float* bias = (const float*)d_in[5];
  float* out = (float*)d_out;

  float* sums = (float*)d_ws;                       // N*F floats
  float* cnt  = sums + (size_t)N_NODES * F_IN;      // N floats

  // 1) zero accumulators
  const int n4 = (N_NODES * F_IN + N_NODES) / 4;    // divisible by 4
  zero_ws_kernel<<<(n4 + 255) / 256, 256, 0, stream>>>((float4*)d_ws, n4);

  // 2) edge messages + scatter add (+degree)
  edge_scatter_kernel<<<N_EDGES / EPB, 256, 0, stream>>>(feat, ew, src, dst, sums, cnt);

  // 3) in-place segment mean
  const int nmean = N_NODES * (F_IN / 4);
  mean_kernel<<<(nmean + 255) / 256, 256, 0, stream>>>((float4*)sums, cnt);

  // 4) fused concat-GEMM + bias + relu (WMMA f32, async LDS staging)
  gemm_wmma_kernel<<<N_NODES / 16, 512, 0, stream>>>(feat, sums, W, bias, out);
}